// MILPCompositeLoss_16793322128029
// MI455X (gfx1250) — compile-verified
//
#include <hip/hip_runtime.h>
#include <math.h>

typedef float v2f __attribute__((ext_vector_type(2)));
typedef float v8f __attribute__((ext_vector_type(8)));

namespace {
constexpr int N_VARS  = 100000;
constexpr int C_CONS  = 50000;
constexpr int E_EDGES = 1600000;
constexpr int K_SOLS  = 16;
constexpr int H_DIM   = 256;
constexpr int M_DIM   = 512;

// workspace layout (float slots)
constexpr int WS_SW     = 0;   // sum of var weights
constexpr int WS_SPW    = 1;   // sum per_var * w
constexpr int WS_SENT   = 2;   // sum entropy (binary mask)
constexpr int WS_CNTB   = 3;   // count of binary vars
constexpr int WS_OBJP   = 4;   // obj_pred
constexpr int WS_SVIOL  = 5;   // sum softplus(5*slack)/5
constexpr int WS_SORTH  = 6;   // sum (gram - I)^2
constexpr int WS_SREC   = 7;   // sum (V_hat - V_micro)^2
constexpr int WS_KBEST  = 8;   // int: argmin(objs)
constexpr int WS_OBJMIN = 9;
constexpr int WS_OBJMAX = 10;
constexpr int WS_XSTE   = 16;                       // N floats
constexpr int WS_AX     = WS_XSTE + N_VARS;         // C floats
constexpr int WS_QN     = WS_AX + C_CONS;           // M*H floats (normalized Q)
constexpr int WS_XMAT   = WS_QN + M_DIM * H_DIM;    // M*H floats (Hn@W^T + b)
}

__device__ __forceinline__ float softplus_f(float x) {
  return log1pf(expf(-fabsf(x))) + fmaxf(x, 0.0f);
}

__device__ __forceinline__ float waveReduceSum(float v) {
  #pragma unroll
  for (int off = 16; off > 0; off >>= 1) v += __shfl_down(v, off, 32);
  return v;
}

// block reduce (blockDim.x multiple of 32, <=256) then one atomicAdd
__device__ __forceinline__ void blockAtomicAdd(float* dst, float v, float* lds8) {
  v = waveReduceSum(v);
  const int lane = threadIdx.x & 31, wave = threadIdx.x >> 5;
  if (lane == 0) lds8[wave] = v;
  __syncthreads();
  if (threadIdx.x == 0) {
    float s = 0.0f;
    const int nw = blockDim.x >> 5;
    for (int i = 0; i < nw; ++i) s += lds8[i];
    atomicAdd(dst, s);
  }
  __syncthreads();
}

// ---------------- zero accumulators + ax buffer ----------------
__global__ void k_zero(float* ws, int n) {
  int i = blockIdx.x * blockDim.x + threadIdx.x;
  if (i < n) ws[i] = 0.0f;
}

// ---------------- objs: argmin / min / max (K=16) ----------------
__global__ void k_objs(const float* objs, float* ws) {
  if (threadIdx.x == 0 && blockIdx.x == 0) {
    float mn = objs[0], mx = objs[0];
    int bi = 0;
    for (int k = 1; k < K_SOLS; ++k) {
      float o = objs[k];
      if (o < mn) { mn = o; bi = k; }
      if (o > mx) mx = o;
    }
    ((int*)ws)[WS_KBEST] = bi;
    ws[WS_OBJMIN] = mn;
    ws[WS_OBJMAX] = mx;
  }
}

// ---------------- loss_sup + entropy + x_ste + obj_pred ----------------
__global__ __launch_bounds__(256)
void k_sup(const float* __restrict__ vf, const float* __restrict__ sols,
           const float* __restrict__ blogit, const float* __restrict__ imu,
           const float* __restrict__ ilstd, const float* __restrict__ cval,
           const float* __restrict__ pred, float* __restrict__ ws) {
  __shared__ float red[8];
  const int j = blockIdx.x * blockDim.x + threadIdx.x;
  float sw = 0.f, spw = 0.f, sent = 0.f, cnt = 0.f, sobj = 0.f;
  if (j < N_VARS) {
    const float f1 = vf[j * 21 + 1];
    const float f2 = vf[j * 21 + 2];
    const float f3 = vf[j * 21 + 3];
    const float f4 = vf[j * 21 + 4];
    const float raw_obj = vf[j * 21 + 20];
    const bool isBin  = (f1 != 0.0f);
    const bool isInt  = (f2 != 0.0f) || (f3 != 0.0f);
    const bool isCont = (f4 != 0.0f);
    const bool binU   = isBin || !(isBin || isInt || isCont);

    float mn = sols[j], mx = sols[j];
    #pragma unroll
    for (int k = 1; k < K_SOLS; ++k) {
      float s = sols[k * N_VARS + j];
      mn = fminf(mn, s); mx = fmaxf(mx, s);
    }
    const int kbest = ((const int*)ws)[WS_KBEST];
    const float best = sols[kbest * N_VARS + j];
    const float range = mx - mn;

    // weight: where(cont, .3, where(int, .8/(1+r), where(binU, 1, 0)))
    float w = isCont ? 0.3f : (isInt ? (0.8f / (1.0f + range)) : (binU ? 1.0f : 0.0f));

    const float logit = blogit[j];
    const float ts = best * (1.0f - 0.01f) + 0.5f * 0.01f;
    const float bce = softplus_f(logit) - logit * ts;
    const float lstd = ilstd[j];
    const float z = (best - imu[j]) * expf(-lstd);
    const float nll = 0.918938533204672742f + lstd + 0.5f * z * z;
    const float dd = cval[j] - best;
    const float mse = dd * dd;
    const float pv = isCont ? mse : (isInt ? nll : (binU ? bce : 0.0f));
    sw = w; spw = pv * w;

    // entropy over binary mask
    float p = 1.0f / (1.0f + expf(-logit));
    p = fminf(fmaxf(p, 1e-6f), 1.0f - 1e-6f);
    const float ent = -(p * logf(p) + (1.0f - p) * logf(1.0f - p));
    if (binU) { sent = ent; cnt = 1.0f; }

    // x_ste = where(cont, pred, where(binU|int, round(pred), 0))
    const float pr = pred[j];
    const float xs = isCont ? pr : ((binU || isInt) ? rintf(pr) : 0.0f);
    ws[WS_XSTE + j] = xs;
    sobj = raw_obj * xs;
  }
  blockAtomicAdd(ws + WS_SW,   sw,   red);
  blockAtomicAdd(ws + WS_SPW,  spw,  red);
  blockAtomicAdd(ws + WS_SENT, sent, red);
  blockAtomicAdd(ws + WS_CNTB, cnt,  red);
  blockAtomicAdd(ws + WS_OBJP, sobj, red);
}

// ---------------- edge scatter: ax[ci] += ev * x_ste[vi] ----------------
__global__ __launch_bounds__(256)
void k_edges(const float* __restrict__ ev, const int* __restrict__ ci,
             const int* __restrict__ vi, float* __restrict__ ws) {
  const int e = blockIdx.x * blockDim.x + threadIdx.x;
  if (e < E_EDGES) {
    const float wx = ev[e] * ws[WS_XSTE + vi[e]];
    atomicAdd(ws + WS_AX + ci[e], wx);
  }
}

// ---------------- violation reduction over constraints ----------------
__global__ __launch_bounds__(256)
void k_viol(const float* __restrict__ con_feats, float* __restrict__ ws) {
  __shared__ float red[8];
  const int c = blockIdx.x * blockDim.x + threadIdx.x;
  float v = 0.0f;
  if (c < C_CONS) {
    const float slack = ws[WS_AX + c] - con_feats[c * 6 + 0];
    v = softplus_f(5.0f * slack) * 0.2f;
  }
  blockAtomicAdd(ws + WS_SVIOL, v, red);
}

// ---------------- normalize rows of Q_macro -> Qn ----------------
__global__ __launch_bounds__(256)
void k_normq(const float* __restrict__ Q, float* __restrict__ ws) {
  __shared__ float red[8];
  __shared__ float s_inv;
  const int m = blockIdx.x, t = threadIdx.x;
  const float q = Q[m * H_DIM + t];
  float s2 = waveReduceSum(q * q);
  const int lane = t & 31, wave = t >> 5;
  if (lane == 0) red[wave] = s2;
  __syncthreads();
  if (t == 0) {
    float s = 0.0f;
    for (int i = 0; i < 8; ++i) s += red[i];
    s_inv = 1.0f / fmaxf(sqrtf(s), 1e-12f);
  }
  __syncthreads();
  ws[WS_QN + m * H_DIM + t] = q * s_inv;
}

// ---------------- loss_orth: mean((Qn Qn^T - I)^2) via WMMA f32 ----------------
__global__ __launch_bounds__(256)
void k_orth(float* __restrict__ ws) {
  __shared__ float Qa[16][H_DIM + 4];
  __shared__ float red[8];
  const float* Qn = ws + WS_QN;
  const int rb = blockIdx.x * 16;
  // stage normalized A rows into LDS
  for (int idx = threadIdx.x; idx < 16 * H_DIM; idx += 256) {
    const int r = idx >> 8, k = idx & (H_DIM - 1);
    Qa[r][k] = Qn[(rb + r) * H_DIM + k];
  }
  __syncthreads();

  const int lane = threadIdx.x & 31, wave = threadIdx.x >> 5;
  const int half = lane >> 4, l = lane & 15;
  float local = 0.0f;
  #pragma unroll
  for (int t = 0; t < 4; ++t) {
    const int cb = (wave * 4 + t) * 16;
    const float* brow = Qn + (cb + l) * H_DIM;  // B[k][n] = Qn[n][k]
    v8f acc = {};
    for (int k0 = 0; k0 < H_DIM; k0 += 4) {
      v2f a, b;
      a.x = Qa[l][k0 + 2 * half];
      a.y = Qa[l][k0 + 2 * half + 1];
      b.x = brow[k0 + 2 * half];
      b.y = brow[k0 + 2 * half + 1];
      acc = __builtin_amdgcn_wmma_f32_16x16x4_f32(
          false, a, false, b, (short)0, acc, false, false);
    }
    #pragma unroll
    for (int r = 0; r < 8; ++r) {
      const int row = rb + r + 8 * half;
      const int col = cb + l;
      const float g = acc[r] - ((row == col) ? 1.0f : 0.0f);
      local += g * g;
    }
  }
  blockAtomicAdd(ws + WS_SORTH, local, red);
}

// ---------------- layernorm(H_macro) @ W^T + b  -> X (fused, per row) ----------------
__global__ __launch_bounds__(256)
void k_ln_x(const float* __restrict__ Hm, const float* __restrict__ W,
            const float* __restrict__ bias, const float* __restrict__ gamma,
            const float* __restrict__ beta, float* __restrict__ ws) {
  __shared__ float hn[H_DIM];
  __shared__ float redA[8], redB[8];
  __shared__ float s_mean, s_inv;
  const int m = blockIdx.x, t = threadIdx.x;
  const float x = Hm[m * H_DIM + t];
  float s1 = waveReduceSum(x);
  float s2 = waveReduceSum(x * x);
  const int lane = t & 31, wave = t >> 5;
  if (lane == 0) { redA[wave] = s1; redB[wave] = s2; }
  __syncthreads();
  if (t == 0) {
    float S = 0.0f, S2 = 0.0f;
    for (int i = 0; i < 8; ++i) { S += redA[i]; S2 += redB[i]; }
    const float mean = S / (float)H_DIM;
    const float var = S2 / (float)H_DIM - mean * mean;
    s_mean = mean;
    s_inv = rsqrtf(var + 1e-5f);
  }
  __syncthreads();
  hn[t] = (x - s_mean) * s_inv * gamma[t] + beta[t];
  __syncthreads();
  // X[m][t] = sum_k hn[k] * W[t][k] + bias[t]
  float acc = 0.0f;
  const float* wr = W + t * H_DIM;
  for (int k = 0; k < H_DIM; ++k) acc = fmaf(hn[k], wr[k], acc);
  ws[WS_XMAT + m * H_DIM + t] = acc + bias[t];
}

// ---------------- loss_recon: sum((A_backward @ X - V_micro)^2) via WMMA f32 ----------------
__global__ __launch_bounds__(256)
void k_recon(const float* __restrict__ A, const float* __restrict__ Vm,
             float* __restrict__ ws) {
  __shared__ float At[16][M_DIM + 4];  // stride 516 -> conflict-free column reads
  __shared__ float red[8];
  const float* X = ws + WS_XMAT;
  const int rb = blockIdx.x * 16;
  // stage 16 x 512 A tile (coalesced along k)
  for (int idx = threadIdx.x; idx < 16 * M_DIM; idx += 256) {
    const int r = idx >> 9, k = idx & (M_DIM - 1);
    At[r][k] = A[(rb + r) * M_DIM + k];
  }
  __syncthreads();

  const int lane = threadIdx.x & 31, wave = threadIdx.x >> 5;
  const int half = lane >> 4, l = lane & 15;
  float local = 0.0f;
  #pragma unroll
  for (int t = 0; t < 2; ++t) {
    const int cb = (wave * 2 + t) * 16;
    v8f acc = {};
    for (int k0 = 0; k0 < M_DIM; k0 += 4) {
      v2f a, b;
      a.x = At[l][k0 + 2 * half];
      a.y = At[l][k0 + 2 * half + 1];
      // B[k][n] = X[k][cb+n], X row-major 512x256, hot in L2 (512 KB)
      b.x = X[(k0 + 2 * half) * H_DIM + cb + l];
      b.y = X[(k0 + 2 * half + 1) * H_DIM + cb + l];
      acc = __builtin_amdgcn_wmma_f32_16x16x4_f32(
          false, a, false, b, (short)0, acc, false, false);
    }
    #pragma unroll
    for (int r = 0; r < 8; ++r) {
      const int row = rb + r + 8 * half;
      const float d = acc[r] - Vm[row * H_DIM + cb + l];
      local += d * d;
    }
  }
  blockAtomicAdd(ws + WS_SREC, local, red);
}

// ---------------- finalize ----------------
__global__ void k_final(const float* __restrict__ ws, float* __restrict__ out) {
  if (threadIdx.x == 0 && blockIdx.x == 0) {
    const float loss_sup = ws[WS_SPW] / (ws[WS_SW] + 1e-8f);
    const float loss_viol = ws[WS_SVIOL] / (float)C_CONS;
    const float obj_range = fmaxf(ws[WS_OBJMAX] - ws[WS_OBJMIN], 1e-6f);
    const float loss_obj = fmaxf((ws[WS_OBJP] - ws[WS_OBJMIN]) / obj_range, 0.0f);
    const float viol = loss_viol + 0.1f * loss_obj;
    const float loss_orth = ws[WS_SORTH] / (float)(M_DIM * M_DIM);
    const float loss_recon = ws[WS_SREC] / (float)((long long)N_VARS * H_DIM);
    const float loss_ent = ws[WS_SENT] / fmaxf(ws[WS_CNTB], 1.0f);
    const float total = 1.0f * loss_sup + 0.1f * viol + 0.01f * loss_orth +
                        0.1f * loss_recon + 0.01f * loss_ent;
    out[0] = total;
    out[1] = loss_sup;
    out[2] = viol;
    out[3] = loss_orth;
    out[4] = loss_recon;
    out[5] = loss_ent;
  }
}

extern "C" void kernel_launch(void* const* d_in, const int* in_sizes, int n_in,
                              void* d_out, int out_size, void* d_ws, size_t ws_size,
                              hipStream_t stream) {
  const float* var_feats  = (const float*)d_in[0];
  const float* con_feats  = (const float*)d_in[1];
  const float* edge_vals  = (const float*)d_in[2];
  const float* sols       = (const float*)d_in[3];
  const float* objs       = (const float*)d_in[4];
  const float* blogit     = (const float*)d_in[5];
  const float* imu        = (const float*)d_in[6];
  const float* ilstd      = (const float*)d_in[7];
  const float* cval       = (const float*)d_in[8];
  const float* pred       = (const float*)d_in[9];
  const float* Q_macro    = (const float*)d_in[10];
  const float* A_backward = (const float*)d_in[11];
  const float* H_macro    = (const float*)d_in[12];
  const float* V_micro    = (const float*)d_in[13];
  const float* W_v_weight = (const float*)d_in[14];
  const float* W_v_bias   = (const float*)d_in[15];
  const float* ln_gamma   = (const float*)d_in[16];
  const float* ln_beta    = (const float*)d_in[17];
  const int*   edge_ci    = (const int*)d_in[18];
  const int*   edge_vi    = (const int*)d_in[19];
  float* ws  = (float*)d_ws;
  float* out = (float*)d_out;

  // zero accumulators + ax (everything before x_ste, plus the ax segment)
  {
    const int nz = WS_XSTE;  // scalar slots
    k_zero<<<(nz + 255) / 256, 256, 0, stream>>>(ws, nz);
    k_zero<<<(C_CONS + 255) / 256, 256, 0, stream>>>(ws + WS_AX, C_CONS);
  }
  k_objs<<<1, 32, 0, stream>>>(objs, ws);
  k_sup<<<(N_VARS + 255) / 256, 256, 0, stream>>>(
      var_feats, sols, blogit, imu, ilstd, cval, pred, ws);
  k_edges<<<(E_EDGES + 255) / 256, 256, 0, stream>>>(edge_vals, edge_ci, edge_vi, ws);
  k_viol<<<(C_CONS + 255) / 256, 256, 0, stream>>>(con_feats, ws);
  k_normq<<<M_DIM, H_DIM, 0, stream>>>(Q_macro, ws);
  k_orth<<<M_DIM / 16, 256, 0, stream>>>(ws);
  k_ln_x<<<M_DIM, H_DIM, 0, stream>>>(H_macro, W_v_weight, W_v_bias, ln_gamma,
                                      ln_beta, ws);
  k_recon<<<N_VARS / 16, 256, 0, stream>>>(A_backward, V_micro, ws);
  k_final<<<1, 1, 0, stream>>>(ws, out);
}